// SnycTwinSSM_67473936220408
// MI455X (gfx1250) — compile-verified
//
#include <hip/hip_runtime.h>

#define LSEQ 2048
#define NB   2
#define DM   512
#define DH   256   // HALF
#define DS   128   // D_STATE
#define DTR  32    // DT_RANK
#define NXD  288   // DTR + 2*DS

typedef float v2f __attribute__((ext_vector_type(2)));
typedef float v8f __attribute__((ext_vector_type(8)));

// CDNA5 fp32 WMMA: D(16x16,f32) = A(16x4,f32) * B(4x16,f32) + C
// A layout: lanes 0-15 hold M=lane {VGPR0:K=0, VGPR1:K=1}; lanes 16-31 {K=2,K=3}
// B layout: VGPR0 {lanes0-15:K=0, lanes16-31:K=2}; VGPR1 {K=1,K=3}; N=lane&15
// C/D layout: VGPR r -> (M=r, N=lane) lanes0-15 ; (M=r+8, N=lane-16) lanes16-31
__device__ __forceinline__ v8f wmma4(v2f a, v2f b, v8f c) {
  return __builtin_amdgcn_wmma_f32_16x16x4_f32(
      /*neg_a=*/false, a, /*neg_b=*/false, b,
      /*c_mod=*/(short)0, c, /*reuse_a=*/false, /*reuse_b=*/false);
}

// ---------------------------------------------------------------------------
// A1 = softthresh(A_base), A2 = softthresh(A1); also init the 4 atomic-max
// slots used by the B/C threshold reductions. Single workgroup (32K elems).
// ---------------------------------------------------------------------------
__global__ __launch_bounds__(1024) void k_prepA(
    const float* __restrict__ A_log, const float* __restrict__ eps,
    float* __restrict__ A1, float* __restrict__ A2,
    unsigned int* __restrict__ slots)
{
  __shared__ float sm[1024];
  const int tid = threadIdx.x;
  const int NTOT = DH * DS;
  const float e = 1.f + eps[0];

  float m = -3.402823466e38f;
  for (int i = tid; i < NTOT; i += 1024)
    m = fmaxf(m, -e * __expf(A_log[i]));
  sm[tid] = m; __syncthreads();
  for (int s = 512; s > 0; s >>= 1) {
    if (tid < s) sm[tid] = fmaxf(sm[tid], sm[tid + s]);
    __syncthreads();
  }
  const float thr1 = 0.1f * sm[0];
  __syncthreads();

  float m2 = -3.402823466e38f;
  for (int i = tid; i < NTOT; i += 1024) {
    const float v   = -e * __expf(A_log[i]);
    const float mag = fmaxf(fabsf(v) - thr1, 0.f);
    const float r   = (v == 0.f) ? 0.f : copysignf(mag, v);
    A1[i] = r;
    m2 = fmaxf(m2, r);
  }
  sm[tid] = m2; __syncthreads();
  for (int s = 512; s > 0; s >>= 1) {
    if (tid < s) sm[tid] = fmaxf(sm[tid], sm[tid + s]);
    __syncthreads();
  }
  const float thr2 = 0.1f * sm[0];
  for (int i = tid; i < NTOT; i += 1024) {
    const float v   = A1[i];
    const float mag = fmaxf(fabsf(v) - thr2, 0.f);
    A2[i] = (v == 0.f) ? 0.f : copysignf(mag, v);
  }
  if (tid < 4) slots[tid] = 0u;   // mapped(-inf) lower bound for atomicMax
}

// ---------------------------------------------------------------------------
// xz = inph @ in_w^T  ->  x_raw/z_raw stored channel-major (B, 256, L)
// One wave computes a 64(M=l) x 32(N=i) tile: 4x2 register blocking so each
// A fragment feeds 2 WMMAs and each B fragment feeds 4 (11 FLOP/byte vs 1).
// ---------------------------------------------------------------------------
__global__ __launch_bounds__(32) void k_gemm_in(
    const float* __restrict__ inp, const float* __restrict__ W,
    float* __restrict__ xr, float* __restrict__ zr, int chanBase)
{
  const int lane = threadIdx.x;
  const int n0 = blockIdx.x << 5;
  const int l0 = blockIdx.y << 6;
  const int b  = blockIdx.z;
  const float* A = inp + ((size_t)b * (2 * DM) + chanBase) * LSEQ;
  const int mA   = l0 + (lane & 15);
  const int kh   = (lane >> 4) << 1;
  const int nIdx = n0 + (lane & 15);
  v8f c[4][2] = {};
  for (int k0 = 0; k0 < DM; k0 += 4) {
    const float* Ak0 = A + (size_t)(k0 + kh) * LSEQ + mA;
    const float* Ak1 = Ak0 + LSEQ;
    v2f a[4], bv[2];
    #pragma unroll
    for (int mt = 0; mt < 4; ++mt) { a[mt].x = Ak0[16 * mt]; a[mt].y = Ak1[16 * mt]; }
    #pragma unroll
    for (int nt = 0; nt < 2; ++nt) {
      const float2 w2 = *(const float2*)(W + (size_t)(nIdx + 16 * nt) * DM + k0 + kh);
      bv[nt].x = w2.x; bv[nt].y = w2.y;
    }
    // WGP-scope prefetch of the strided A stream 8 K-steps ahead
    if (k0 + 8 < DM) __builtin_prefetch(Ak0 + (size_t)8 * LSEQ, 0, 3);
    #pragma unroll
    for (int mt = 0; mt < 4; ++mt)
      #pragma unroll
      for (int nt = 0; nt < 2; ++nt)
        c[mt][nt] = wmma4(a[mt], bv[nt], c[mt][nt]);
  }
  const int moff = (lane >> 4) << 3;
  #pragma unroll
  for (int nt = 0; nt < 2; ++nt) {
    const int i = nIdx + 16 * nt;
    float* dstrow = (i < DH)
        ? xr + ((size_t)b * DH + i) * LSEQ
        : zr + ((size_t)b * DH + (i - DH)) * LSEQ;
    #pragma unroll
    for (int mt = 0; mt < 4; ++mt) {
      float* dst = dstrow + l0 + 16 * mt + moff;
      *(float4*)dst       = make_float4(c[mt][nt][0], c[mt][nt][1], c[mt][nt][2], c[mt][nt][3]);
      *(float4*)(dst + 4) = make_float4(c[mt][nt][4], c[mt][nt][5], c[mt][nt][6], c[mt][nt][7]);
    }
  }
}

// depthwise conv k=4 (pad left 1, right 2) + SiLU, (B,256,L) -> (B,256,L)
__global__ __launch_bounds__(256) void k_conv_silu(
    const float* __restrict__ src, const float* __restrict__ w,
    float* __restrict__ dst)
{
  const int idx = blockIdx.x * 256 + threadIdx.x;   // NB*DH*LSEQ threads
  const int l  = idx & (LSEQ - 1);
  const int ch = (idx >> 11) & (DH - 1);
  const float* row = src + (idx - l);
  const float4 wc = *(const float4*)(w + ch * 4);
  float acc = wc.y * row[l];
  if (l >= 1)       acc += wc.x * row[l - 1];
  if (l + 1 < LSEQ) acc += wc.z * row[l + 1];
  if (l + 2 < LSEQ) acc += wc.w * row[l + 2];
  dst[idx] = acc / (1.f + __expf(-acc));   // silu
}

// x_dbl = xc^T @ xproj_w^T -> stored row-major (B, L, 288); 64x32 wave tiles
__global__ __launch_bounds__(32) void k_gemm_xproj(
    const float* __restrict__ xc, const float* __restrict__ W,
    float* __restrict__ xdbl)
{
  const int lane = threadIdx.x;
  const int n0 = blockIdx.x << 5;   // N = 288 -> 9 tiles of 32
  const int l0 = blockIdx.y << 6;
  const int b  = blockIdx.z;
  const float* A = xc + (size_t)b * DH * LSEQ;
  const int mA   = l0 + (lane & 15);
  const int kh   = (lane >> 4) << 1;
  const int nIdx = n0 + (lane & 15);
  v8f c[4][2] = {};
  for (int k0 = 0; k0 < DH; k0 += 4) {
    const float* Ak0 = A + (size_t)(k0 + kh) * LSEQ + mA;
    const float* Ak1 = Ak0 + LSEQ;
    v2f a[4], bv[2];
    #pragma unroll
    for (int mt = 0; mt < 4; ++mt) { a[mt].x = Ak0[16 * mt]; a[mt].y = Ak1[16 * mt]; }
    #pragma unroll
    for (int nt = 0; nt < 2; ++nt) {
      const float2 w2 = *(const float2*)(W + (size_t)(nIdx + 16 * nt) * DH + k0 + kh);
      bv[nt].x = w2.x; bv[nt].y = w2.y;
    }
    #pragma unroll
    for (int mt = 0; mt < 4; ++mt)
      #pragma unroll
      for (int nt = 0; nt < 2; ++nt)
        c[mt][nt] = wmma4(a[mt], bv[nt], c[mt][nt]);
  }
  const int moff = (lane >> 4) << 3;
  #pragma unroll
  for (int mt = 0; mt < 4; ++mt) {
    float* base = xdbl + ((size_t)b * LSEQ + l0 + 16 * mt + moff) * NXD + nIdx;
    #pragma unroll
    for (int r = 0; r < 8; ++r) {
      base[(size_t)r * NXD]      = c[mt][0][r];
      base[(size_t)r * NXD + 16] = c[mt][1][r];
    }
  }
}

// global max over a 128-wide column slice of x_dbl (monotone uint mapping)
__global__ __launch_bounds__(256) void k_maxreduce(
    const float* __restrict__ xdbl, int jbase, unsigned int* __restrict__ slot)
{
  __shared__ unsigned int sm[256];
  unsigned int best = 0u;
  const int total = NB * LSEQ * DS;
  for (int idx = blockIdx.x * 256 + threadIdx.x; idx < total;
       idx += gridDim.x * 256) {
    const int n   = idx & (DS - 1);
    const int row = idx >> 7;
    unsigned int u = __float_as_uint(xdbl[(size_t)row * NXD + jbase + n]);
    u = (u & 0x80000000u) ? ~u : (u | 0x80000000u);
    best = best > u ? best : u;
  }
  sm[threadIdx.x] = best; __syncthreads();
  for (int s = 128; s > 0; s >>= 1) {
    if (threadIdx.x < s) sm[threadIdx.x] = max(sm[threadIdx.x], sm[threadIdx.x + s]);
    __syncthreads();
  }
  if (threadIdx.x == 0) atomicMax(slot, sm[0]);
}

// in-place soft-threshold: v -> sign(v)*relu(|v| - 0.1*max)
__global__ __launch_bounds__(256) void k_softthresh(
    float* __restrict__ xdbl, int jbase, const unsigned int* __restrict__ slot)
{
  const unsigned int u = *slot;
  const float mx  = __uint_as_float((u & 0x80000000u) ? (u & 0x7FFFFFFFu) : ~u);
  const float thr = 0.1f * mx;
  const int idx = blockIdx.x * 256 + threadIdx.x;
  const int n   = idx & (DS - 1);
  const int row = idx >> 7;
  float* p = xdbl + (size_t)row * NXD + jbase + n;
  const float v   = *p;
  const float mag = fmaxf(fabsf(v) - thr, 0.f);
  *p = (v == 0.f) ? 0.f : copysignf(mag, v);
}

// delta = softplus(x_dbl[:, :32] @ dt_w^T + dt_b) -> (B, 256, L); 64x32 tiles
__global__ __launch_bounds__(32) void k_gemm_dt(
    const float* __restrict__ xdbl, const float* __restrict__ W,
    const float* __restrict__ bias, float* __restrict__ delta)
{
  const int lane = threadIdx.x;
  const int n0 = blockIdx.x << 5;
  const int l0 = blockIdx.y << 6;
  const int b  = blockIdx.z;
  const int kh   = (lane >> 4) << 1;
  const int nIdx = n0 + (lane & 15);
  const float* Arow[4];
  #pragma unroll
  for (int mt = 0; mt < 4; ++mt)
    Arow[mt] = xdbl + ((size_t)b * LSEQ + l0 + 16 * mt + (lane & 15)) * NXD;
  v8f c[4][2] = {};
  for (int k0 = 0; k0 < DTR; k0 += 4) {
    v2f a[4], bv[2];
    #pragma unroll
    for (int mt = 0; mt < 4; ++mt) {
      const float2 a2 = *(const float2*)(Arow[mt] + k0 + kh);
      a[mt].x = a2.x; a[mt].y = a2.y;
    }
    #pragma unroll
    for (int nt = 0; nt < 2; ++nt) {
      const float2 w2 = *(const float2*)(W + (size_t)(nIdx + 16 * nt) * DTR + k0 + kh);
      bv[nt].x = w2.x; bv[nt].y = w2.y;
    }
    #pragma unroll
    for (int mt = 0; mt < 4; ++mt)
      #pragma unroll
      for (int nt = 0; nt < 2; ++nt)
        c[mt][nt] = wmma4(a[mt], bv[nt], c[mt][nt]);
  }
  const int moff = (lane >> 4) << 3;
  #pragma unroll
  for (int nt = 0; nt < 2; ++nt) {
    const float bb = bias[nIdx + 16 * nt];
    float* dstrow = delta + ((size_t)b * DH + nIdx + 16 * nt) * LSEQ;
    #pragma unroll
    for (int mt = 0; mt < 4; ++mt) {
      float o[8];
      #pragma unroll
      for (int r = 0; r < 8; ++r) {
        const float x = c[mt][nt][r] + bb;
        o[r] = fmaxf(x, 0.f) + log1pf(__expf(-fabsf(x)));   // stable softplus
      }
      float* dst = dstrow + l0 + 16 * mt + moff;
      *(float4*)dst       = make_float4(o[0], o[1], o[2], o[3]);
      *(float4*)(dst + 4) = make_float4(o[4], o[5], o[6], o[7]);
    }
  }
}

// selective scan: one wave32 per (b,d) row, 4 states per lane (n = 4*lane+j).
// Delta/u fetched as coalesced 32-wide chunks and broadcast via shfl; y
// reduced with a 5-step shfl_xor butterfly and stored coalesced per chunk.
__global__ __launch_bounds__(256) void k_scan(
    const float* __restrict__ delta, const float* __restrict__ u,
    const float* __restrict__ Amat, const float* __restrict__ xdbl,
    const float* __restrict__ Dvec, float* __restrict__ y)
{
  const int gtid = blockIdx.x * 256 + threadIdx.x;
  const int wave = gtid >> 5;
  const int lane = threadIdx.x & 31;
  const int b = wave >> 8;
  const int d = wave & (DH - 1);
  const float4 a4 = *(const float4*)(Amat + d * DS + lane * 4);
  const float Dd = Dvec[d];
  const float* drow = delta + ((size_t)b * DH + d) * LSEQ;
  const float* urow = u     + ((size_t)b * DH + d) * LSEQ;
  const float* xb   = xdbl  + (size_t)b * LSEQ * NXD;
  float* yrow = y + ((size_t)b * DH + d) * LSEQ;
  float hx = 0.f, hy = 0.f, hz = 0.f, hw = 0.f;
  for (int l0 = 0; l0 < LSEQ; l0 += 32) {
    const float dt32 = drow[l0 + lane];   // coalesced chunk loads
    const float uu32 = urow[l0 + lane];
    float ybuf = 0.f;
    for (int j = 0; j < 32; ++j) {
      const float dt = __shfl(dt32, j, 32);
      const float uu = __shfl(uu32, j, 32);
      const float* col = xb + (size_t)(l0 + j) * NXD;
      const float4 bv = *(const float4*)(col + DTR + lane * 4);        // Bc
      const float4 cv = *(const float4*)(col + DTR + DS + lane * 4);   // Cc
      const float du = dt * uu;
      hx = __expf(dt * a4.x) * hx + du * bv.x;
      hy = __expf(dt * a4.y) * hy + du * bv.y;
      hz = __expf(dt * a4.z) * hz + du * bv.z;
      hw = __expf(dt * a4.w) * hw + du * bv.w;
      float acc = hx * cv.x + hy * cv.y + hz * cv.z + hw * cv.w;
      #pragma unroll
      for (int off = 16; off > 0; off >>= 1) acc += __shfl_xor(acc, off, 32);
      if (j == lane) ybuf = acc + uu * Dd;
    }
    yrow[l0 + lane] = ybuf;   // coalesced chunk store
  }
}

// out = [y ; z]^T @ out_w^T -> (B, L, 512) row-major; 64x32 wave tiles
__global__ __launch_bounds__(32) void k_gemm_out(
    const float* __restrict__ yb, const float* __restrict__ zc,
    const float* __restrict__ W, float* __restrict__ out)
{
  const int lane = threadIdx.x;
  const int n0 = blockIdx.x << 5;
  const int l0 = blockIdx.y << 6;
  const int b  = blockIdx.z;
  const int mA   = l0 + (lane & 15);
  const int kh   = (lane >> 4) << 1;
  const int nIdx = n0 + (lane & 15);
  const float* Ay = yb + (size_t)b * DH * LSEQ;
  const float* Az = zc + (size_t)b * DH * LSEQ;
  v8f c[4][2] = {};
  for (int k0 = 0; k0 < DM; k0 += 4) {
    const float* Ah = (k0 < DH) ? Ay : Az;   // uniform across wave
    const int kk = k0 & (DH - 1);
    const float* Ak0 = Ah + (size_t)(kk + kh) * LSEQ + mA;
    const float* Ak1 = Ak0 + LSEQ;
    v2f a[4], bv[2];
    #pragma unroll
    for (int mt = 0; mt < 4; ++mt) { a[mt].x = Ak0[16 * mt]; a[mt].y = Ak1[16 * mt]; }
    #pragma unroll
    for (int nt = 0; nt < 2; ++nt) {
      const float2 w2 = *(const float2*)(W + (size_t)(nIdx + 16 * nt) * DM + k0 + kh);
      bv[nt].x = w2.x; bv[nt].y = w2.y;
    }
    // WGP-scope prefetch of the strided A stream 8 K-steps ahead
    if ((k0 & (DH - 1)) + 8 < DH) __builtin_prefetch(Ak0 + (size_t)8 * LSEQ, 0, 3);
    #pragma unroll
    for (int mt = 0; mt < 4; ++mt)
      #pragma unroll
      for (int nt = 0; nt < 2; ++nt)
        c[mt][nt] = wmma4(a[mt], bv[nt], c[mt][nt]);
  }
  const int moff = (lane >> 4) << 3;
  #pragma unroll
  for (int mt = 0; mt < 4; ++mt) {
    float* base = out + ((size_t)b * LSEQ + l0 + 16 * mt + moff) * DM + nIdx;
    #pragma unroll
    for (int r = 0; r < 8; ++r) {
      base[(size_t)r * DM]      = c[mt][0][r];
      base[(size_t)r * DM + 16] = c[mt][1][r];
    }
  }
}

extern "C" void kernel_launch(void* const* d_in, const int* in_sizes, int n_in,
                              void* d_out, int out_size, void* d_ws, size_t ws_size,
                              hipStream_t stream)
{
  (void)in_sizes; (void)n_in; (void)out_size; (void)ws_size;
  const float* inp   = (const float*)d_in[0];
  const float* A_log = (const float*)d_in[15];
  const float* Dvec  = (const float*)d_in[16];
  const float* eps   = (const float*)d_in[17];

  // workspace carve (~30 MB, reused across the two sequential branches)
  float* ws = (float*)d_ws;
  float* A1 = ws;                           ws += DH * DS;
  float* A2 = ws;                           ws += DH * DS;
  unsigned int* slots = (unsigned int*)ws;  ws += 16;
  float* x_raw = ws;  ws += (size_t)NB * DH * LSEQ;
  float* z_raw = ws;  ws += (size_t)NB * DH * LSEQ;
  float* xcb   = ws;  ws += (size_t)NB * DH * LSEQ;
  float* zcb   = ws;  ws += (size_t)NB * DH * LSEQ;
  float* xdbl  = ws;  ws += (size_t)NB * LSEQ * NXD;
  float* delta = ws;  ws += (size_t)NB * DH * LSEQ;
  float* ybuf  = ws;  ws += (size_t)NB * DH * LSEQ;

  k_prepA<<<1, 1024, 0, stream>>>(A_log, eps, A1, A2, slots);

  for (int br = 0; br < 2; ++br) {
    const float* in_w = (const float*)d_in[br ? 8 : 1];
    const float* cxw  = (const float*)d_in[br ? 9 : 2];
    const float* czw  = (const float*)d_in[br ? 10 : 3];
    const float* xpw  = (const float*)d_in[br ? 11 : 4];
    const float* dtw  = (const float*)d_in[br ? 12 : 5];
    const float* dtb  = (const float*)d_in[br ? 13 : 6];
    const float* outw = (const float*)d_in[br ? 14 : 7];
    const float* Ab   = br ? A2 : A1;
    unsigned int* slotB = slots + br * 2;
    unsigned int* slotC = slotB + 1;
    float* outp = (float*)d_out + (size_t)br * NB * LSEQ * DM;

    k_gemm_in<<<dim3(DM / 32, LSEQ / 64, NB), 32, 0, stream>>>(
        inp, in_w, x_raw, z_raw, br * DM);
    const int nconv = NB * DH * LSEQ;
    k_conv_silu<<<nconv / 256, 256, 0, stream>>>(x_raw, cxw, xcb);
    k_conv_silu<<<nconv / 256, 256, 0, stream>>>(z_raw, czw, zcb);
    k_gemm_xproj<<<dim3(NXD / 32, LSEQ / 64, NB), 32, 0, stream>>>(
        xcb, xpw, xdbl);
    k_maxreduce<<<512, 256, 0, stream>>>(xdbl, DTR, slotB);
    k_maxreduce<<<512, 256, 0, stream>>>(xdbl, DTR + DS, slotC);
    const int nth = NB * LSEQ * DS;
    k_softthresh<<<nth / 256, 256, 0, stream>>>(xdbl, DTR, slotB);
    k_softthresh<<<nth / 256, 256, 0, stream>>>(xdbl, DTR + DS, slotC);
    k_gemm_dt<<<dim3(DH / 32, LSEQ / 64, NB), 32, 0, stream>>>(
        xdbl, dtw, dtb, delta);
    k_scan<<<(NB * DH * 32) / 256, 256, 0, stream>>>(
        delta, xcb, Ab, xdbl, Dvec, ybuf);
    k_gemm_out<<<dim3(DM / 32, LSEQ / 64, NB), 32, 0, stream>>>(
        ybuf, zcb, outw, outp);
  }
}